// ColdDiffusion_44470091382916
// MI455X (gfx1250) — compile-verified
//
#include <hip/hip_runtime.h>

typedef __attribute__((ext_vector_type(2))) float v2f;
typedef __attribute__((ext_vector_type(4))) float v4f;
typedef __attribute__((ext_vector_type(8))) float v8f;

// D = A(16x4, f32) * B(4x16, f32) + C(16x16, f32)
// 8-arg form: (neg_a, A, neg_b, B, c_mod, C, reuse_a, reuse_b); f32 A/B have no neg -> false.
#define WMMA_F32_16X16X4(A, Bm, Cm) \
  __builtin_amdgcn_wmma_f32_16x16x4_f32(false, (A), false, (Bm), (short)0, (Cm), false, false)

// Problem shape (fixed by the reference)
constexpr int kNB    = 16;          // batch
constexpr int kNN    = 2048;        // points per batch
constexpr int kM     = 8192;        // anchors
constexpr int kD     = 8;           // feature dim (4x2 flattened)
constexpr int kRows  = kNB * kNN;   // 32768 x-rows
constexpr int kThreads = 256;       // 8 waves per block
constexpr int kBlocks  = kRows / 16 / 8; // 2048 row-tiles / 8 waves = 256 blocks

__global__ __launch_bounds__(kThreads)
void cold_diffusion_qsample_kernel(const float* __restrict__ x,       // [32768][8]
                                   const float* __restrict__ anchors, // [8192][8]
                                   const float* __restrict__ sqa,     // [1000]
                                   const float* __restrict__ sqb,     // [1000]
                                   const int*   __restrict__ t,       // [16]
                                   float*       __restrict__ out)     // [32768][8]
{
    // LDS: all anchors (256KB) + their squared norms (32KB) = 288KB of the WGP's 320KB
    __shared__ float ldsA[kM * kD];
    __shared__ float ldsN[kM];

    const int tid = threadIdx.x;

    // ---- Stage anchors + norms into LDS (uniform loop, no divergence) ----
    for (int m = tid; m < kM; m += kThreads) {
        v4f a0 = *(const v4f*)(anchors + m * kD);
        v4f a1 = *(const v4f*)(anchors + m * kD + 4);
        *(v4f*)(ldsA + m * kD)     = a0;
        *(v4f*)(ldsA + m * kD + 4) = a1;
        float nrm = a0.x*a0.x + a0.y*a0.y + a0.z*a0.z + a0.w*a0.w
                  + a1.x*a1.x + a1.y*a1.y + a1.z*a1.z + a1.w*a1.w;
        ldsN[m] = nrm;
    }
    __syncthreads();

    const int lane   = tid & 31;
    const int wave   = tid >> 5;
    const int half   = lane >> 4;        // 0: lanes 0-15, 1: lanes 16-31
    const int l16    = lane & 15;
    const int k0     = half * 2;         // K-pair this lane supplies
    const int rowBase = (blockIdx.x * 8 + wave) * 16;

    // ---- B fragment: x' = -2*x for 16 rows, K=8 split into two 4xK fragments.
    // B 4x16 f32 layout: VGPR0 lanes0-15 = K=0, lanes16-31 = K=2; VGPR1 = K=1 / K=3.
    const float* xr = x + (size_t)(rowBase + l16) * kD;
    v2f bLo, bHi;
    {
        v2f xl = *(const v2f*)(xr + k0);
        v2f xh = *(const v2f*)(xr + k0 + 4);
        bLo.x = -2.0f * xl.x;  bLo.y = -2.0f * xl.y;
        bHi.x = -2.0f * xh.x;  bHi.y = -2.0f * xh.y;
    }

    // ---- Sweep all 512 anchor tiles.
    // 4 independent argmin chains per lane (chain j tracks accumulator rows r=j and r=j+4)
    // so the cmp/cndmask work pipelines instead of forming one serial dependency chain.
    float bv0 = 3.402823466e+38f, bv1 = 3.402823466e+38f;
    float bv2 = 3.402823466e+38f, bv3 = 3.402823466e+38f;
    int   bi0 = 0, bi1 = 0, bi2 = 0, bi3 = 0;

    for (int tb = 0; tb < kM; tb += 16) {
        // A fragment (16 anchors x K=8): lane supplies anchors[tb + l16][k0..k0+1] (lo), +4 (hi)
        const float* ar = ldsA + (size_t)(tb + l16) * kD;
        v2f aLo = *(const v2f*)(ar + k0);
        v2f aHi = *(const v2f*)(ar + k0 + 4);

        // C = ||a||^2 broadcast per D-row: lane's VGPR r holds anchor m = tb + 8*half + r
        v8f c = *(const v8f*)(ldsN + tb + 8 * half);

        c = WMMA_F32_16X16X4(aLo, bLo, c);   // accumulate K=0..3
        c = WMMA_F32_16X16X4(aHi, bHi, c);   // accumulate K=4..7

        const int ibase = tb + 8 * half;
        // Indices within each chain are strictly increasing across the sweep,
        // so strict < keeps the lowest index on ties (reference argmin semantics).
        if (c[0] < bv0) { bv0 = c[0]; bi0 = ibase + 0; }
        if (c[1] < bv1) { bv1 = c[1]; bi1 = ibase + 1; }
        if (c[2] < bv2) { bv2 = c[2]; bi2 = ibase + 2; }
        if (c[3] < bv3) { bv3 = c[3]; bi3 = ibase + 3; }
        if (c[4] < bv0) { bv0 = c[4]; bi0 = ibase + 4; }
        if (c[5] < bv1) { bv1 = c[5]; bi1 = ibase + 5; }
        if (c[6] < bv2) { bv2 = c[6]; bi2 = ibase + 6; }
        if (c[7] < bv3) { bv3 = c[7]; bi3 = ibase + 7; }
    }

    // ---- Merge the 4 chains with full (val, idx) tie-break -> lowest index wins ----
    if (bv1 < bv0 || (bv1 == bv0 && bi1 < bi0)) { bv0 = bv1; bi0 = bi1; }
    if (bv3 < bv2 || (bv3 == bv2 && bi3 < bi2)) { bv2 = bv3; bi2 = bi3; }
    if (bv2 < bv0 || (bv2 == bv0 && bi2 < bi0)) { bv0 = bv2; bi0 = bi2; }

    // ---- Merge the two half-wave anchor subsets (lane L <-> L^16), tie -> lower index ----
    float ov = __shfl_xor(bv0, 16, 32);
    int   oi = __shfl_xor(bi0, 16, 32);
    if (ov < bv0 || (ov == bv0 && oi < bi0)) { bv0 = ov; bi0 = oi; }

    // ---- Epilogue: out = sa*x + sb*anchors[argmin]; lanes 0-15 each own one row ----
    if (half == 0) {
        const int row = rowBase + l16;
        const int b   = row >> 11;         // row / 2048
        const int tt  = t[b];
        const float sa = sqa[tt];
        const float sb = sqb[tt];

        const float* xo = x + (size_t)row * kD;
        const float* ao = ldsA + (size_t)bi0 * kD;

        v4f x0 = *(const v4f*)(xo);
        v4f x1 = *(const v4f*)(xo + 4);
        v4f a0 = *(const v4f*)(ao);
        v4f a1 = *(const v4f*)(ao + 4);

        v4f o0, o1;
        o0.x = sa * x0.x + sb * a0.x;  o0.y = sa * x0.y + sb * a0.y;
        o0.z = sa * x0.z + sb * a0.z;  o0.w = sa * x0.w + sb * a0.w;
        o1.x = sa * x1.x + sb * a1.x;  o1.y = sa * x1.y + sb * a1.y;
        o1.z = sa * x1.z + sb * a1.z;  o1.w = sa * x1.w + sb * a1.w;

        *(v4f*)(out + (size_t)row * kD)     = o0;
        *(v4f*)(out + (size_t)row * kD + 4) = o1;
    }
}

extern "C" void kernel_launch(void* const* d_in, const int* in_sizes, int n_in,
                              void* d_out, int out_size, void* d_ws, size_t ws_size,
                              hipStream_t stream) {
    (void)in_sizes; (void)n_in; (void)out_size; (void)d_ws; (void)ws_size;
    const float* x       = (const float*)d_in[0];
    const float* anchors = (const float*)d_in[1];
    const float* sqa     = (const float*)d_in[2];
    const float* sqb     = (const float*)d_in[3];
    const int*   t       = (const int*)d_in[4];
    float* out = (float*)d_out;

    cold_diffusion_qsample_kernel<<<dim3(kBlocks), dim3(kThreads), 0, stream>>>(
        x, anchors, sqa, sqb, t, out);
}